// CGA_model_57724360458771
// MI455X (gfx1250) — compile-verified
//
#include <hip/hip_runtime.h>
#include <hip/hip_bf16.h>
#include <math.h>

// -------- WMMA vector types (wave32, gfx1250) --------
typedef __attribute__((ext_vector_type(2))) float v2f; // A/B of 16x16x4 f32 (2 VGPRs/lane)
typedef __attribute__((ext_vector_type(8))) float v8f; // C/D 16x16 f32 (8 VGPRs/lane)

// ---------------------------------------------------------------------------
// zero a float buffer (float4-wide)
__global__ __launch_bounds__(256) void zero_f4(float4* __restrict__ p, int n4) {
    int i = blockIdx.x * blockDim.x + threadIdx.x;
    if (i < n4) p[i] = make_float4(0.f, 0.f, 0.f, 0.f);
}

// h = x * node_imp (row scale), d multiple of 4
__global__ __launch_bounds__(256) void scale_rows(const float* __restrict__ x,
                                                  const float* __restrict__ imp,
                                                  float* __restrict__ h,
                                                  int n, int d) {
    int chunks = d >> 2;
    int i = blockIdx.x * blockDim.x + threadIdx.x;
    if (i >= n * chunks) return;
    int node = i / chunks;
    int c = (i - node * chunks) << 2;
    float s = imp[node];
    float4 v = *(const float4*)(x + (size_t)node * d + c);
    v.x *= s; v.y *= s; v.z *= s; v.w *= s;
    *(float4*)(h + (size_t)node * d + c) = v;
}

// agg[dst] += relu(h[src] * ew)   -- L2-resident gather + hardware f32 atomics
__global__ __launch_bounds__(256) void scatter_msg(const float* __restrict__ h,
                                                   const int* __restrict__ src,
                                                   const int* __restrict__ dst,
                                                   const float* __restrict__ ew,
                                                   float* __restrict__ agg,
                                                   int d, int nedges) {
    int e = blockIdx.x * blockDim.x + threadIdx.x;
    if (e >= nedges) return;
    int c = blockIdx.y << 2;  // feature chunk of 4
    float w = ew[e];
    int s = src[e], t = dst[e];
    float4 v = *(const float4*)(h + (size_t)s * d + c);
    float* ap = agg + (size_t)t * d + c;
    unsafeAtomicAdd(ap + 0, fmaxf(v.x * w, 0.f));
    unsafeAtomicAdd(ap + 1, fmaxf(v.y * w, 0.f));
    unsafeAtomicAdd(ap + 2, fmaxf(v.z * w, 0.f));
    unsafeAtomicAdd(ap + 3, fmaxf(v.w * w, 0.f));
}

// ---------------------------------------------------------------------------
// Fused WMMA GEMM:  out = act( BN( (alpha*A0 [+ A1]) @ W + bias ) )
//   HAS_A1 (compile-time): fuse z = (1+eps)*A0 + A1 into the A-fragment load.
//   A0[,A1]: [nrows,K] row-major; W: [K,OutC] row-major;
//   bias: [OutC]; bn: [4,OutC] {gamma,beta,mean,var} folded to scale/shift.
//   4 waves/block; each wave -> 32 rows x 64 cols:
//     8 v8f accumulators, every B fragment feeds 2 WMMAs (B reuse x2),
//     8 independent WMMAs per k-step (no RAW hazards between them).
// Requires: nrows % 32 == 0, OutC % 64 == 0, K % 4 == 0.
// All WMMA-executing control flow is wave-uniform -> EXEC all-ones (ISA 7.12).
template <bool HAS_A1>
__global__ __launch_bounds__(128) void gemm_wmma_f32(
    const float* __restrict__ A0, const float* __restrict__ A1,
    const float* __restrict__ epsPtr,
    const float* __restrict__ W, const float* __restrict__ bias,
    const float* __restrict__ bn,
    float* __restrict__ out,
    int nrows, int K, int OutC, int act)
{
    const int lane = threadIdx.x & 31;
    const int wave = threadIdx.x >> 5;
    const int m    = lane & 15;
    const int half = (lane >> 4) & 1;
    const int rg   = blockIdx.x * 4 + wave;      // 32-row group (wave-uniform)
    if (rg * 32 >= nrows) return;                // uniform across the wave
    const int row0 = rg * 32 + m;                // rows m and m+16 of the group
    const int row1 = row0 + 16;
    const int cb0  = blockIdx.y * 64;

    const float alpha = HAS_A1 ? (1.0f + epsPtr[0]) : 1.0f;

    // Strength-reduced pointers: bump instead of re-deriving addresses.
    const float* __restrict__ a0p = A0 + (size_t)row0 * K + 2 * half;
    const float* __restrict__ a0q = A0 + (size_t)row1 * K + 2 * half;
    const float* __restrict__ a1p = HAS_A1 ? (A1 + (size_t)row0 * K + 2 * half) : nullptr;
    const float* __restrict__ a1q = HAS_A1 ? (A1 + (size_t)row1 * K + 2 * half) : nullptr;
    const float* __restrict__ wp  = W + (size_t)(2 * half) * OutC + cb0 + m;
    const size_t wstep = (size_t)4 * OutC;

    v8f acc[8] = {};                             // [colTile j][rowTile r] = acc[2j+r]

    for (int kk = 0; kk < K; kk += 4) {
        v2f a0, a1;
        a0.x = a0p[0]; a0.y = a0p[1];
        a1.x = a0q[0]; a1.y = a0q[1];
        if (HAS_A1) {                            // z = (1+eps)*h + agg (branch-free)
            a0.x = alpha * a0.x + a1p[0];
            a0.y = alpha * a0.y + a1p[1];
            a1.x = alpha * a1.x + a1q[0];
            a1.y = alpha * a1.y + a1q[1];
            a1p += 4; a1q += 4;
        }
        v2f b0, b1, b2, b3;
        b0.x = wp[0];  b0.y = wp[OutC + 0];
        b1.x = wp[16]; b1.y = wp[OutC + 16];
        b2.x = wp[32]; b2.y = wp[OutC + 32];
        b3.x = wp[48]; b3.y = wp[OutC + 48];

        acc[0] = __builtin_amdgcn_wmma_f32_16x16x4_f32(false, a0, false, b0, (short)0, acc[0], false, false);
        acc[1] = __builtin_amdgcn_wmma_f32_16x16x4_f32(false, a1, false, b0, (short)0, acc[1], false, false);
        acc[2] = __builtin_amdgcn_wmma_f32_16x16x4_f32(false, a0, false, b1, (short)0, acc[2], false, false);
        acc[3] = __builtin_amdgcn_wmma_f32_16x16x4_f32(false, a1, false, b1, (short)0, acc[3], false, false);
        acc[4] = __builtin_amdgcn_wmma_f32_16x16x4_f32(false, a0, false, b2, (short)0, acc[4], false, false);
        acc[5] = __builtin_amdgcn_wmma_f32_16x16x4_f32(false, a1, false, b2, (short)0, acc[5], false, false);
        acc[6] = __builtin_amdgcn_wmma_f32_16x16x4_f32(false, a0, false, b3, (short)0, acc[6], false, false);
        acc[7] = __builtin_amdgcn_wmma_f32_16x16x4_f32(false, a1, false, b3, (short)0, acc[7], false, false);

        a0p += 4; a0q += 4; wp += wstep;
    }

#pragma unroll
    for (int j = 0; j < 4; ++j) {
        const int col = cb0 + j * 16 + m;        // this lane's output column
        const float bs = bias[col];
        const float g  = bn[col];
        const float be = bn[OutC + col];
        const float mu = bn[2 * OutC + col];
        const float va = bn[3 * OutC + col];
        const float scale = g * rsqrtf(va + 1e-5f);
        const float shift = be - mu * scale;
#pragma unroll
        for (int v = 0; v < 8; ++v) {
            const int r0 = rg * 32 + v + 8 * half;        // D layout: M = v + 8*half
            float v0 = (acc[2 * j + 0][v] + bs) * scale + shift;
            float v1 = (acc[2 * j + 1][v] + bs) * scale + shift;
            if (act) { v0 = fmaxf(v0, 0.f); v1 = fmaxf(v1, 0.f); }
            out[(size_t)r0 * OutC + col]        = v0;
            out[(size_t)(r0 + 16) * OutC + col] = v1;
        }
    }
}

// ---------------------------------------------------------------------------
// node_key = sigmoid(h @ att_W + att_b); one wave32 per node
__global__ __launch_bounds__(256) void att_kernel(const float* __restrict__ h,
                                                  const float* __restrict__ attW,
                                                  const float* __restrict__ attb,
                                                  float* __restrict__ nk, int n, int d) {
    int w = (blockIdx.x * blockDim.x + threadIdx.x) >> 5;
    int lane = threadIdx.x & 31;
    if (w >= n) return;
    const float* hr = h + (size_t)w * d;
    float s = 0.f;
    for (int i = lane; i < d; i += 32) s += hr[i] * attW[i];
#pragma unroll
    for (int off = 16; off > 0; off >>= 1) s += __shfl_xor(s, off, 32);
    if (lane == 0) {
        float z = s + attb[0];
        nk[w] = 1.0f / (1.0f + __expf(-z));
    }
}

// edge_key = node_key[src] * node_key[dst]
__global__ __launch_bounds__(256) void edge_kernel(const float* __restrict__ nk,
                                                   const int* __restrict__ src,
                                                   const int* __restrict__ dst,
                                                   float* __restrict__ ek, int nedges) {
    int e = blockIdx.x * blockDim.x + threadIdx.x;
    if (e < nedges) ek[e] = nk[src[e]] * nk[dst[e]];
}

// ---------------------------------------------------------------------------
static void run_conv(const float* hin, int din,
                     float* agg, float* t, float* hout,
                     const float* W1, const float* b1, const float* bn1,
                     const float* W2, const float* b2, const float* obn,
                     const float* epsPtr, int act_last,
                     const int* src, const int* dst, const float* ew,
                     int n, int nedges, hipStream_t stream)
{
    // zero agg
    int n4 = (n * din) >> 2;
    zero_f4<<<(n4 + 255) / 256, 256, 0, stream>>>((float4*)agg, n4);
    // scatter messages
    dim3 sg((nedges + 255) / 256, din >> 2);
    scatter_msg<<<sg, 256, 0, stream>>>(hin, src, dst, ew, agg, din, nedges);
    // GEMM1: t = relu(bn1(((1+eps)*h + agg) @ W1 + b1))   [n, 512]
    dim3 g1((n / 32 + 3) / 4, 512 / 64);
    gemm_wmma_f32<true><<<g1, 128, 0, stream>>>(hin, agg, epsPtr, W1, b1, bn1, t, n, din, 512, 1);
    // GEMM2: hout = act(obn(t @ W2 + b2))                 [n, 256]
    dim3 g2((n / 32 + 3) / 4, 256 / 64);
    gemm_wmma_f32<false><<<g2, 128, 0, stream>>>(t, nullptr, nullptr, W2, b2, obn, hout, n, 512, 256, act_last);
}

extern "C" void kernel_launch(void* const* d_in, const int* in_sizes, int n_in,
                              void* d_out, int out_size, void* d_ws, size_t ws_size,
                              hipStream_t stream) {
    const float* x        = (const float*)d_in[0];
    const int*   ei       = (const int*)d_in[1];
    const float* node_imp = (const float*)d_in[2];
    const float* edge_imp = (const float*)d_in[3];
    const float* c0_W1    = (const float*)d_in[4];
    const float* c0_b1    = (const float*)d_in[5];
    const float* c0_bn    = (const float*)d_in[6];
    const float* c0_W2    = (const float*)d_in[7];
    const float* c0_b2    = (const float*)d_in[8];
    const float* c0_eps   = (const float*)d_in[9];
    const float* cs_W1    = (const float*)d_in[10];
    const float* cs_b1    = (const float*)d_in[11];
    const float* cs_bn    = (const float*)d_in[12];
    const float* cs_W2    = (const float*)d_in[13];
    const float* cs_b2    = (const float*)d_in[14];
    const float* cs_eps   = (const float*)d_in[15];
    const float* obn      = (const float*)d_in[16];
    const float* att_W    = (const float*)d_in[17];
    const float* att_b    = (const float*)d_in[18];

    const int n      = in_sizes[2];   // 20000 nodes
    const int nedges = in_sizes[3];   // 320000 edges
    const int IN_DIM = 128, EMBD = 256;
    const int* src = ei;
    const int* dst = ei + nedges;

    float* ws  = (float*)d_ws;
    float* h   = ws;                               // [n,256] (conv0 uses stride 128)
    float* agg = ws + (size_t)n * EMBD;            // [n,256]
    float* t   = ws + (size_t)n * (2 * EMBD);      // [n,512]
    float* out = (float*)d_out;                    // node_key [n] ++ edge_key [E]

    // h0 = x * node_imp  ([n,128])
    {
        int total = n * (IN_DIM >> 2);
        scale_rows<<<(total + 255) / 256, 256, 0, stream>>>(x, node_imp, h, n, IN_DIM);
    }

    // conv0: 128 -> 512 -> 256, obn[0], relu
    run_conv(h, IN_DIM, agg, t, h,
             c0_W1, c0_b1, c0_bn, c0_W2, c0_b2, obn + 0 * 4 * EMBD,
             c0_eps, /*act_last=*/1, src, dst, edge_imp, n, nedges, stream);

    // conv1: 256 -> 512 -> 256, obn[1], relu  (cs index 0)
    run_conv(h, EMBD, agg, t, h,
             cs_W1 + 0 * EMBD * 512, cs_b1 + 0 * 512, cs_bn + 0 * 4 * 512,
             cs_W2 + 0 * 512 * EMBD, cs_b2 + 0 * EMBD, obn + 1 * 4 * EMBD,
             cs_eps + 0, /*act_last=*/1, src, dst, edge_imp, n, nedges, stream);

    // conv2: 256 -> 512 -> 256, obn[2], NO relu  (cs index 1)
    run_conv(h, EMBD, agg, t, h,
             cs_W1 + 1 * EMBD * 512, cs_b1 + 1 * 512, cs_bn + 1 * 4 * 512,
             cs_W2 + 1 * 512 * EMBD, cs_b2 + 1 * EMBD, obn + 2 * 4 * EMBD,
             cs_eps + 1, /*act_last=*/0, src, dst, edge_imp, n, nedges, stream);

    // node_key = sigmoid(h @ att_W + att_b) -> out[0..n)
    att_kernel<<<(n * 32 + 255) / 256, 256, 0, stream>>>(h, att_W, att_b, out, n, EMBD);
    // edge_key = nk[src]*nk[dst] -> out[n..n+E)
    edge_kernel<<<(nedges + 255) / 256, 256, 0, stream>>>(out, src, dst, out + n, nedges);
}